// BiLSTMClassifier_24850680775002
// MI455X (gfx1250) — compile-verified
//
#include <hip/hip_runtime.h>
#include <math.h>

// Problem constants
#define BB 256
#define TT 512
#define DD 128
#define UU 32
#define CC 5
#define BT (BB * TT)   // 131072 rows

typedef float v2f __attribute__((ext_vector_type(2)));
typedef float v8f __attribute__((ext_vector_type(8)));
typedef int   v4i __attribute__((ext_vector_type(4)));

#define GAS __attribute__((address_space(1)))
#define LAS __attribute__((address_space(3)))

// Feature probes (device pass only; host pass falls back harmlessly).
#if __has_builtin(__builtin_amdgcn_global_load_async_to_lds_b128) && \
    __has_builtin(__builtin_amdgcn_s_wait_asynccnt)
#define HAVE_ASYNC_LDS 1
#else
#define HAVE_ASYNC_LDS 0
#endif

#if __has_builtin(__builtin_amdgcn_tanhf)
#define HAVE_HW_TANH 1
#else
#define HAVE_HW_TANH 0
#endif

// fp32 WMMA: D(16x16 f32) = A(16x4 f32) * B(4x16 f32) + C
__device__ __forceinline__ v8f wmma_f32(v2f a, v2f b, v8f c) {
    return __builtin_amdgcn_wmma_f32_16x16x4_f32(false, a, false, b, (short)0, c,
                                                 false, false);
}

__device__ __forceinline__ float fast_tanh(float x) {
#if HAVE_HW_TANH
    return __builtin_amdgcn_tanhf(x);   // v_tanh_f32 (gfx1250 TRANS op)
#else
    return tanhf(x);
#endif
}

__device__ __forceinline__ float sigm(float x) {
    // sigmoid(x) = 0.5 + 0.5*tanh(0.5*x): one TRANS op + 2 VALU
    return fmaf(0.5f, fast_tanh(0.5f * x), 0.5f);
}

#if HAVE_ASYNC_LDS
// Async global -> LDS copy of 16 bytes (GLOBAL_LOAD_ASYNC_TO_LDS_B128, ASYNCcnt).
__device__ __forceinline__ void async_b128(const float* g, float* l) {
    __builtin_amdgcn_global_load_async_to_lds_b128(
        (GAS v4i*)(GAS void*)g, (LAS v4i*)(LAS void*)l, 0, 0);
}
#endif

// ---------------------------------------------------------------------------
// Kernel 1: layer-1 input projections, both directions.
// xp1[dir][row, n] = sum_k x[row,k] * K1[dir][k,n] + bias[dir][0][n]
// rows = B*T (131072), k = 128, n = 96.  One wave per (dir, 16-row tile).
// ---------------------------------------------------------------------------
__global__ __launch_bounds__(128) void proj1_kernel(
    const float* __restrict__ x,
    const float* __restrict__ k1f, const float* __restrict__ b1f,
    const float* __restrict__ k1b, const float* __restrict__ b1b,
    float* __restrict__ xp1f, float* __restrict__ xp1b) {
    int wave = (blockIdx.x << 2) | (threadIdx.x >> 5);   // 16384 waves
    int lane = threadIdx.x & 31;
    int l16 = lane & 15, half = lane >> 4;
    int dir = wave & 1;
    int m0 = (wave >> 1) << 4;

    const float* K  = dir ? k1b : k1f;          // [128, 96] row-major
    const float* Bi = dir ? b1b : b1f;          // bias[0] = first 96 floats
    float* out = dir ? xp1b : xp1f;

    v8f acc[6] = {};                            // 6 n-tiles of 16
    const float* arow = x + (size_t)(m0 + l16) * DD + 2 * half;

#pragma unroll 4
    for (int k0 = 0; k0 < DD; k0 += 4) {
        v2f a;
        a.x = arow[k0];
        a.y = arow[k0 + 1];
        const float* brow = K + (size_t)(k0 + 2 * half) * 96 + l16;
#pragma unroll
        for (int j = 0; j < 6; ++j) {
            v2f b;
            b.x = brow[16 * j];
            b.y = brow[96 + 16 * j];
            acc[j] = wmma_f32(a, b, acc[j]);
        }
    }
#pragma unroll
    for (int j = 0; j < 6; ++j) {
        float bias = Bi[16 * j + l16];
#pragma unroll
        for (int r = 0; r < 8; ++r) {
            out[(size_t)(m0 + r + 8 * half) * 96 + 16 * j + l16] = acc[j][r] + bias;
        }
    }
}

// ---------------------------------------------------------------------------
// Kernel 2: layer-1 GRU recurrence (tanh). One wave per (dir, 16-batch tile):
// 32 blocks x 32 threads. h (16x32) lives in LDS; recurrent weights (32x96)
// held as persistent WMMA B-fragments in VGPRs; per step: [16,32]@[32,96]
// via 8 k-steps x 6 n-tiles of V_WMMA_F32_16X16X4_F32, then gate math.
// xp gate inputs are double-buffered in LDS via async global->LDS loads
// (ASYNCcnt) issued one timestep ahead to hide L2 latency.
// ---------------------------------------------------------------------------
__global__ __launch_bounds__(32) void gru1_kernel(
    const float* __restrict__ xp1f, const float* __restrict__ xp1b,
    const float* __restrict__ r1f, const float* __restrict__ b1f,
    const float* __restrict__ r1b, const float* __restrict__ b1b,
    float* __restrict__ h1f, float* __restrict__ h1b) {
    // h state: stride 36 so the two half-waves hit disjoint LDS bank groups
    // (8*36 == 288 == 32 mod 64) for both A-fragment and gate accesses.
    __shared__ __align__(16) float hsh[16][36];
#if HAVE_ASYNC_LDS
    // xp staging: 2 slots x 16 rows x 96 cols, row stride 100 (800 == 32 mod 64).
    __shared__ __align__(16) float xbuf[2][16][100];
#endif
    int lane = threadIdx.x & 31;
    int l16 = lane & 15, half = lane >> 4;
    int dir = blockIdx.x & 1;
    int b0 = (blockIdx.x >> 1) << 4;

    const float* XP = dir ? xp1b : xp1f;
    const float* R  = dir ? r1b : r1f;            // [32, 96]
    const float* RB = (dir ? b1b : b1f) + 96;     // bias[1]
    float* H = dir ? h1b : h1f;

    // Persistent recurrent-weight B fragments: 8 k-steps x 6 n-tiles
    v2f bf[8][6];
#pragma unroll
    for (int ks = 0; ks < 8; ++ks) {
        const float* brow = R + (size_t)(4 * ks + 2 * half) * 96 + l16;
#pragma unroll
        for (int j = 0; j < 6; ++j) {
            bf[ks][j].x = brow[16 * j];
            bf[ks][j].y = brow[96 + 16 * j];
        }
    }
    float rbv[6];
#pragma unroll
    for (int j = 0; j < 6; ++j) rbv[j] = RB[16 * j + l16];

    for (int i = lane; i < 16 * 36; i += 32) (&hsh[0][0])[i] = 0.0f;
    __syncthreads();

#if HAVE_ASYNC_LDS
    // Per-lane prefetch plan: 384 float4 per step, 12 per lane.
    const float* gbase[12];
    int loff[12];
#pragma unroll
    for (int k = 0; k < 12; ++k) {
        int i4 = lane + 32 * k;          // 0..383
        int row = i4 / 24;               // 24 float4 per 96-float row
        int c4 = i4 - row * 24;
        gbase[k] = XP + (size_t)(b0 + row) * TT * 96 + c4 * 4;
        loff[k] = row * 100 + c4 * 4;    // floats within a slot
    }
    // Prologue: prefetch timestep 0 into slot 0.
    {
        int t0 = dir ? (TT - 1) : 0;
        size_t toff = (size_t)t0 * 96;
        float* slot = &xbuf[0][0][0];
#pragma unroll
        for (int k = 0; k < 12; ++k) async_b128(gbase[k] + toff, slot + loff[k]);
    }
#endif

    for (int t = 0; t < TT; ++t) {
        int tcur = dir ? (TT - 1 - t) : t;

#if HAVE_ASYNC_LDS
        // Issue next step's prefetch, then wait for this step's 12 loads
        // (async loads complete in order; <=12 outstanding leaves only the
        // new batch in flight).
        if (t + 1 < TT) {
            int tnext = dir ? (TT - 2 - t) : (t + 1);
            size_t toff = (size_t)tnext * 96;
            float* slot = &xbuf[(t + 1) & 1][0][0];
#pragma unroll
            for (int k = 0; k < 12; ++k) async_b128(gbase[k] + toff, slot + loff[k]);
            __builtin_amdgcn_s_wait_asynccnt(12);
        } else {
            __builtin_amdgcn_s_wait_asynccnt(0);
        }
        const float (*xs)[100] = xbuf[t & 1];
#endif

        v8f acc[6] = {};
#pragma unroll
        for (int ks = 0; ks < 8; ++ks) {
            v2f a;
            a.x = hsh[l16][4 * ks + 2 * half];
            a.y = hsh[l16][4 * ks + 2 * half + 1];
#pragma unroll
            for (int j = 0; j < 6; ++j) acc[j] = wmma_f32(a, bf[ks][j], acc[j]);
        }
        // Gate math: each lane owns 16 unique (m,u) cells.
#pragma unroll
        for (int j = 0; j < 2; ++j) {
            int u = 16 * j + l16;
#pragma unroll
            for (int r = 0; r < 8; ++r) {
                int m = r + 8 * half;
#if HAVE_ASYNC_LDS
                float xz = xs[m][u];
                float xr = xs[m][32 + u];
                float xh = xs[m][64 + u];
#else
                size_t base = ((size_t)(b0 + m) * TT + tcur) * 96;
                float xz = XP[base + u];
                float xr = XP[base + 32 + u];
                float xh = XP[base + 64 + u];
#endif
                float z  = sigm(xz + acc[j][r] + rbv[j]);
                float rg = sigm(xr + acc[j + 2][r] + rbv[j + 2]);
                float hh = fast_tanh(xh + rg * (acc[j + 4][r] + rbv[j + 4]));
                float hold = hsh[m][u];
                float hnew = z * hold + (1.0f - z) * hh;
                hsh[m][u] = hnew;
                H[((size_t)(b0 + m) * TT + tcur) * UU + u] = hnew;
            }
        }
        __syncthreads();
    }
}

// ---------------------------------------------------------------------------
// Kernel 3: layer-2 input projections. h = h1f + h1b summed on the fly;
// xp2[dir] = h @ k2[dir] (32 x 15, padded to 16) + bias. One wave / 16 rows.
// ---------------------------------------------------------------------------
__global__ __launch_bounds__(128) void proj2_kernel(
    const float* __restrict__ h1f, const float* __restrict__ h1b,
    const float* __restrict__ k2f, const float* __restrict__ b2f,
    const float* __restrict__ k2b, const float* __restrict__ b2b,
    float* __restrict__ xp2f, float* __restrict__ xp2b) {
    int wave = (blockIdx.x << 2) | (threadIdx.x >> 5);   // 8192 waves
    int lane = threadIdx.x & 31;
    int l16 = lane & 15, half = lane >> 4;
    int m0 = wave << 4;

    v8f accf = {}, accb = {};
    const float* af = h1f + (size_t)(m0 + l16) * UU + 2 * half;
    const float* ab = h1b + (size_t)(m0 + l16) * UU + 2 * half;
    bool valid = (l16 < 15);

#pragma unroll
    for (int ks = 0; ks < 8; ++ks) {
        v2f a;
        a.x = af[4 * ks] + ab[4 * ks];
        a.y = af[4 * ks + 1] + ab[4 * ks + 1];
        int kk = 4 * ks + 2 * half;
        v2f bfv, bbv;
        bfv.x = valid ? k2f[kk * 15 + l16] : 0.0f;
        bfv.y = valid ? k2f[(kk + 1) * 15 + l16] : 0.0f;
        bbv.x = valid ? k2b[kk * 15 + l16] : 0.0f;
        bbv.y = valid ? k2b[(kk + 1) * 15 + l16] : 0.0f;
        accf = wmma_f32(a, bfv, accf);
        accb = wmma_f32(a, bbv, accb);
    }
    float biasf = valid ? b2f[l16] : 0.0f;   // bias[0] of (2,15)
    float biasb = valid ? b2b[l16] : 0.0f;
#pragma unroll
    for (int r = 0; r < 8; ++r) {
        size_t o = (size_t)(m0 + r + 8 * half) * 16 + l16;
        xp2f[o] = accf[r] + biasf;
        xp2b[o] = accb[r] + biasb;
    }
}

// ---------------------------------------------------------------------------
// Kernel 4: layer-2 GRU recurrence with softmax candidate activation.
// One thread per (dir, batch): 512 threads, h[5] in registers.
// ---------------------------------------------------------------------------
__global__ __launch_bounds__(32) void gru2_kernel(
    const float* __restrict__ xp2f, const float* __restrict__ xp2b,
    const float* __restrict__ r2f, const float* __restrict__ b2f,
    const float* __restrict__ r2b, const float* __restrict__ b2b,
    float* __restrict__ emf, float* __restrict__ emb) {
    int g = blockIdx.x * 32 + threadIdx.x;   // 0..511
    int dir = g & 1;
    int b = g >> 1;

    const float* XP = dir ? xp2b : xp2f;
    const float* R  = dir ? r2b : r2f;            // [5, 15]
    const float* RB = (dir ? b2b : b2f) + 15;     // bias[1]
    float* EM = dir ? emb : emf;

    float R2[CC][15], RBv[15];
#pragma unroll
    for (int c = 0; c < CC; ++c)
#pragma unroll
        for (int j = 0; j < 15; ++j) R2[c][j] = R[c * 15 + j];
#pragma unroll
    for (int j = 0; j < 15; ++j) RBv[j] = RB[j];

    float h[CC] = {0, 0, 0, 0, 0};
    for (int t = 0; t < TT; ++t) {
        int tcur = dir ? (TT - 1 - t) : t;
        const float* xp = XP + ((size_t)b * TT + tcur) * 16;
        float hrk[15];
#pragma unroll
        for (int j = 0; j < 15; ++j) {
            float s = RBv[j];
#pragma unroll
            for (int c = 0; c < CC; ++c) s += h[c] * R2[c][j];
            hrk[j] = s;
        }
        float z[CC], cand[CC];
        float mx = -1e30f;
#pragma unroll
        for (int c = 0; c < CC; ++c) {
            z[c] = sigm(xp[c] + hrk[c]);
            float rg = sigm(xp[CC + c] + hrk[CC + c]);
            cand[c] = xp[2 * CC + c] + rg * hrk[2 * CC + c];
            mx = fmaxf(mx, cand[c]);
        }
        float ssum = 0.0f;
#pragma unroll
        for (int c = 0; c < CC; ++c) {
            cand[c] = __expf(cand[c] - mx);
            ssum += cand[c];
        }
        float inv = 1.0f / ssum;
#pragma unroll
        for (int c = 0; c < CC; ++c) {
            float hh = cand[c] * inv;
            h[c] = z[c] * h[c] + (1.0f - z[c]) * hh;
            EM[((size_t)b * TT + tcur) * CC + c] = h[c];
        }
    }
}

// ---------------------------------------------------------------------------
// Kernel 5: CRF Viterbi decode. One thread per batch element.
// ---------------------------------------------------------------------------
__global__ __launch_bounds__(32) void viterbi_kernel(
    const float* __restrict__ emf, const float* __restrict__ emb,
    const float* __restrict__ trans,
    unsigned char* __restrict__ bp,
    int* __restrict__ tags, float* __restrict__ score) {
    int b = blockIdx.x * 32 + threadIdx.x;   // 0..255
    float TR[CC][CC];
#pragma unroll
    for (int p = 0; p < CC; ++p)
#pragma unroll
        for (int c = 0; c < CC; ++c) TR[p][c] = trans[p * CC + c];

    size_t e0 = (size_t)b * TT * CC;
    float alpha[CC];
#pragma unroll
    for (int c = 0; c < CC; ++c) alpha[c] = emf[e0 + c] + emb[e0 + c];

    unsigned char* mybp = bp + (size_t)b * (TT - 1) * CC;
    for (int t = 1; t < TT; ++t) {
        size_t et = e0 + (size_t)t * CC;
        float na[CC];
#pragma unroll
        for (int c = 0; c < CC; ++c) {
            float best = alpha[0] + TR[0][c];
            int arg = 0;
#pragma unroll
            for (int p = 1; p < CC; ++p) {
                float s = alpha[p] + TR[p][c];
                if (s > best) { best = s; arg = p; }   // first-max like argmax
            }
            na[c] = best + emf[et + c] + emb[et + c];
            mybp[(size_t)(t - 1) * CC + c] = (unsigned char)arg;
        }
#pragma unroll
        for (int c = 0; c < CC; ++c) alpha[c] = na[c];
    }
    float best = alpha[0];
    int last = 0;
#pragma unroll
    for (int c = 1; c < CC; ++c)
        if (alpha[c] > best) { best = alpha[c]; last = c; }

    int* tg = tags + (size_t)b * TT;
    tg[TT - 1] = last;
    int cur = last;
    for (int t = TT - 2; t >= 0; --t) {
        cur = mybp[(size_t)t * CC + cur];
        tg[t] = cur;
    }
    score[b] = best;
}

// ---------------------------------------------------------------------------
extern "C" void kernel_launch(void* const* d_in, const int* in_sizes, int n_in,
                              void* d_out, int out_size, void* d_ws, size_t ws_size,
                              hipStream_t stream) {
    (void)in_sizes; (void)n_in; (void)out_size; (void)ws_size;
    const float* x     = (const float*)d_in[0];
    const float* k1f   = (const float*)d_in[1];
    const float* r1f   = (const float*)d_in[2];
    const float* b1f   = (const float*)d_in[3];
    const float* k1b   = (const float*)d_in[4];
    const float* r1b   = (const float*)d_in[5];
    const float* b1b   = (const float*)d_in[6];
    const float* k2f   = (const float*)d_in[7];
    const float* r2f   = (const float*)d_in[8];
    const float* b2f   = (const float*)d_in[9];
    const float* k2b   = (const float*)d_in[10];
    const float* r2b   = (const float*)d_in[11];
    const float* b2b   = (const float*)d_in[12];
    const float* trans = (const float*)d_in[13];

    // Workspace layout (floats), with region reuse. Peak = BT*256 floats (128 MiB).
    float* w    = (float*)d_ws;
    float* xp1f = w;                                // [BT, 96]
    float* xp1b = w + (size_t)BT * 96;              // [BT, 96]
    float* h1f  = w + (size_t)BT * 192;             // [BT, 32]
    float* h1b  = h1f + (size_t)BT * 32;            // [BT, 32]
    // After gru1 consumes xp1*, reuse that region:
    float* xp2f = w;                                // [BT, 16]
    float* xp2b = xp2f + (size_t)BT * 16;           // [BT, 16]
    float* emf  = xp2b + (size_t)BT * 16;           // [BT, 5]
    float* emb  = emf + (size_t)BT * CC;            // [BT, 5]
    // After proj2/gru2 consume h1*, reuse that region for backpointers:
    unsigned char* bp = (unsigned char*)h1f;        // [B, T-1, 5] bytes

    int*   tags  = (int*)d_out;                     // [B, T] int32
    float* score = (float*)d_out + (size_t)BB * TT; // [B] f32

    proj1_kernel<<<4096, 128, 0, stream>>>(x, k1f, b1f, k1b, b1b, xp1f, xp1b);
    gru1_kernel<<<32, 32, 0, stream>>>(xp1f, xp1b, r1f, b1f, r1b, b1b, h1f, h1b);
    proj2_kernel<<<2048, 128, 0, stream>>>(h1f, h1b, k2f, b2f, k2b, b2b, xp2f, xp2b);
    gru2_kernel<<<16, 32, 0, stream>>>(xp2f, xp2b, r2f, b2f, r2b, b2b, emf, emb);
    viterbi_kernel<<<8, 32, 0, stream>>>(emf, emb, trans, bp, tags, score);
}